// DNM_Conv_76888504533026
// MI455X (gfx1250) — compile-verified
//
#include <hip/hip_runtime.h>
#include <math.h>

typedef __attribute__((ext_vector_type(2))) float v2f;
typedef __attribute__((ext_vector_type(8))) float v8f;

#define BATCH 64
#define NN    196   // n (contraction dim K)
#define DD    768   // d (GEMM N)
#define OO    196   // o (GEMM M)
#define OPAD  208   // o padded to 13*16
#define MT    13    // OPAD/16
#define NT    48    // DD/16
#define NTG   6     // NT/8  (8 N-tiles per block, one per wave)
#define KSTEPS 49   // NN/4
#define NTILE 7     // ceil(NN/32) n-tiles for the LN+transpose kernel

// workspace layout in floats
#define WP_OFF 0                  // Wp[OPAD][NN]  (zero-padded rows >= 196)
#define C_OFF  (OPAD * NN)        // c[OPAD]
#define YT_OFF (C_OFF + 256)      // yT[BATCH][DD][NN]  (transposed LN output)

// ---------------------------------------------------------------------------
// Kernel 1: fold dendrite weights into synapse weights.
//   Wp[o][n] = dw[o][0]*sw[o][0][n] + dw[o][1]*sw[o][1][n]
//   c[o]     = dw[o][0]*sum_n sb[o][0][n] + dw[o][1]*sum_n sb[o][1][n]
// ---------------------------------------------------------------------------
__global__ void dnm_prep_kernel(const float* __restrict__ sw,
                                const float* __restrict__ sb,
                                const float* __restrict__ dw,
                                float* __restrict__ ws) {
  __shared__ float red[NN];
  const int o = blockIdx.x;   // 0..OPAD-1
  const int t = threadIdx.x;  // 0..NN-1
  float* Wp = ws + WP_OFF;
  float* cv = ws + C_OFF;
  if (o < OO) {
    const float d0 = dw[o * 2 + 0];
    const float d1 = dw[o * 2 + 1];
    Wp[(size_t)o * NN + t] = d0 * sw[(size_t)(o * 2 + 0) * NN + t] +
                             d1 * sw[(size_t)(o * 2 + 1) * NN + t];
    red[t] = d0 * sb[(size_t)(o * 2 + 0) * NN + t] +
             d1 * sb[(size_t)(o * 2 + 1) * NN + t];
    __syncthreads();
    if (t == 0) {
      float s = 0.0f;
      for (int i = 0; i < NN; ++i) s += red[i];
      cv[o] = s;
    }
  } else {
    Wp[(size_t)o * NN + t] = 0.0f;  // zero-pad rows 196..207
    if (t == 0) cv[o] = 0.0f;
  }
}

// ---------------------------------------------------------------------------
// Kernel 2: LayerNorm over d=768 fused with transpose to yT[b][d][n].
// One block per (b, 32-row n-tile). Stats per row via wave32 shuffles, then
// 32x32 LDS tile transpose so both x reads and yT writes are coalesced.
// ---------------------------------------------------------------------------
__global__ void __launch_bounds__(256) dnm_ln_t_kernel(const float* __restrict__ x,
                                                       const float* __restrict__ lw,
                                                       const float* __restrict__ lb,
                                                       float* __restrict__ ws) {
  const int b  = blockIdx.x / NTILE;
  const int n0 = (blockIdx.x % NTILE) * 32;
  const int t    = threadIdx.x;
  const int lane = t & 31;
  const int w    = t >> 5;
  float* yT = ws + YT_OFF;

  __shared__ float mus[32], rss[32];
  __shared__ float tile[32][33];  // +1 pad: no LDS bank conflicts on transpose

  // --- stats: wave w handles rows n0 + 4w .. n0 + 4w+3 (coalesced x reads)
  for (int rr = 0; rr < 4; ++rr) {
    const int r   = w * 4 + rr;
    const int row = n0 + r;
    if (row < NN) {
      const float* xr = x + ((size_t)b * NN + row) * DD;
      float s = 0.0f, q = 0.0f;
      for (int j = lane; j < DD; j += 32) {
        const float v = xr[j];
        s += v;
        q += v * v;
      }
      for (int off = 16; off > 0; off >>= 1) {
        s += __shfl_xor(s, off, 32);
        q += __shfl_xor(q, off, 32);
      }
      if (lane == 0) {
        const float mu  = s * (1.0f / DD);
        const float var = q * (1.0f / DD) - mu * mu;
        mus[r] = mu;
        rss[r] = rsqrtf(var + 1e-5f);
      }
    }
  }
  __syncthreads();

  // --- normalize + transpose, 32x32 chunks marching over d
  for (int dc = 0; dc < DD / 32; ++dc) {
    const int d0 = dc * 32;
#pragma unroll
    for (int p = 0; p < 4; ++p) {
      const int r   = (t >> 5) + p * 8;  // local n row
      const int c   = t & 31;            // local d col (coalesced read)
      const int row = n0 + r;
      float v = 0.0f;
      if (row < NN) {
        const float xv = x[((size_t)b * NN + row) * DD + d0 + c];
        v = (xv - mus[r]) * rss[r] * lw[d0 + c] + lb[d0 + c];
      }
      tile[r][c] = v;
    }
    __syncthreads();
#pragma unroll
    for (int p = 0; p < 4; ++p) {
      const int dr = (t >> 5) + p * 8;   // local d row
      const int nc = t & 31;             // local n col (coalesced write)
      if (n0 + nc < NN)
        yT[((size_t)b * DD + d0 + dr) * NN + n0 + nc] = tile[nc][dr];
    }
    __syncthreads();
  }
}

// ---------------------------------------------------------------------------
// Kernel 3: batched GEMM t[b] = Wp (OPADxNN) x yT[b]^T via
// V_WMMA_F32_16X16X4_F32. Both A and B fragments are per-lane contiguous
// b64 streams in K. Fused bias + exact GELU + x*(out+1) epilogue.
// Block = 8 waves sharing one (b, M-tile); wave w takes N-tile ntg*8+w.
// ---------------------------------------------------------------------------
__global__ void __launch_bounds__(256) dnm_gemm_kernel(const float* __restrict__ x,
                                                       const float* __restrict__ ws,
                                                       float* __restrict__ out) {
  const float* Wp = ws + WP_OFF;
  const float* cv = ws + C_OFF;
  const float* yT = ws + YT_OFF;

  const int wv   = threadIdx.x >> 5;
  const int lane = threadIdx.x & 31;

  const int b   = blockIdx.x / (MT * NTG);
  const int r   = blockIdx.x % (MT * NTG);
  const int mt  = r / NTG;
  const int ntg = r % NTG;
  const int nt  = ntg * 8 + wv;

  const int mo  = mt * 16;                // o tile base (padded space)
  const int col = nt * 16 + (lane & 15);  // global d column
  const int hi  = lane >> 4;              // 0: K=0,1   1: K=2,3 (ISA layout)

  // A: Wp row (mo + lane&15), contiguous in K.  B: yT row for column `col`,
  // contiguous in K. Both 8-byte aligned (NN*4 = 784 = 16-byte multiple).
  const float* Arow = Wp + (size_t)(mo + (lane & 15)) * NN + hi * 2;
  const float* Brow = yT + ((size_t)b * DD + col) * NN + hi * 2;

  // Prefetch the epilogue x cacheline while the WMMA chain runs.
  {
    const int opf = (mo + hi * 8 < OO) ? (mo + hi * 8) : (OO - 1);
    __builtin_prefetch(&x[((size_t)b * OO + opf) * DD + col], 0, 3);
  }

  v8f acc = {};
#pragma unroll 7
  for (int kb = 0; kb < KSTEPS; ++kb) {
    const v2f a  = *(const v2f*)(Arow + kb * 4);
    const v2f bf = *(const v2f*)(Brow + kb * 4);
    acc = __builtin_amdgcn_wmma_f32_16x16x4_f32(
        /*neg_a=*/false, a, /*neg_b=*/false, bf,
        /*c_mod=*/(short)0, acc, /*reuse_a=*/false, /*reuse_b=*/false);
  }

  // Epilogue: C/D layout VGPR v -> M = v + 8*hi, N = lane&15.
#pragma unroll
  for (int v = 0; v < 8; ++v) {
    const int o = mo + hi * 8 + v;
    if (o < OO) {
      const float tv = acc[v] + cv[o];
      const float g  = 0.5f * tv * (1.0f + erff(tv * 0.70710678118654752f));
      const size_t idx = ((size_t)b * OO + o) * DD + col;
      out[idx] = x[idx] * (g + 1.0f);
    }
  }
}

// ---------------------------------------------------------------------------
extern "C" void kernel_launch(void* const* d_in, const int* in_sizes, int n_in,
                              void* d_out, int out_size, void* d_ws, size_t ws_size,
                              hipStream_t stream) {
  const float* x    = (const float*)d_in[0];
  const float* ln_w = (const float*)d_in[1];
  const float* ln_b = (const float*)d_in[2];
  const float* sw   = (const float*)d_in[3];
  const float* sb   = (const float*)d_in[4];
  const float* dw   = (const float*)d_in[5];
  float* out = (float*)d_out;
  float* ws  = (float*)d_ws;

  dnm_prep_kernel<<<dim3(OPAD), dim3(NN), 0, stream>>>(sw, sb, dw, ws);
  dnm_ln_t_kernel<<<dim3(BATCH * NTILE), dim3(256), 0, stream>>>(x, ln_w, ln_b, ws);
  dnm_gemm_kernel<<<dim3(BATCH * MT * NTG), dim3(256), 0, stream>>>(x, ws, out);
}